// SlaterPooling_45543833207162
// MI455X (gfx1250) — compile-verified
//
#include <hip/hip_runtime.h>
#include <hip/hip_bf16.h>

#define NUP_   16
#define NMO_   64
#define NCONF_ 128
#define NSAMP_ 2048

typedef float v2f  __attribute__((ext_vector_type(2)));
typedef float v8f  __attribute__((ext_vector_type(8)));
typedef unsigned int u32x4 __attribute__((ext_vector_type(4)));
typedef int   i32x4 __attribute__((ext_vector_type(4)));
typedef int   i32x8 __attribute__((ext_vector_type(8)));

// One block = one sample s. 256 threads = 8 waves; wave w handles configs
// [w*16, w*16+16). Per config: gather-by-WMMA (exact: selection matrix is 0/1),
// then dual shuffle-LU (lanes 0-15: up, lanes 16-31: down).
__global__ __launch_bounds__(256) void slater_det_kernel(
    const float* __restrict__ x,     // (NSAMP, 32, 64) f32
    const int*   __restrict__ cup,   // (NCONF, 16) i32
    const int*   __restrict__ cdn,   // (NCONF, 16) i32
    float*       __restrict__ out)   // (NSAMP, NCONF) f32
{
    __shared__ float lds_x[32 * NMO_];          // 8 KB: this sample's 32x64 block
    __shared__ float lds_mat[8][2][16][17];     // per-wave scratch, stride-17 rows

    const int s    = blockIdx.x;
    const int tid  = threadIdx.x;
    const int wave = tid >> 5;
    const int lane = tid & 31;
    const int row  = lane & 15;
    const int hh   = lane >> 4;        // 0: lanes 0-15, 1: lanes 16-31
    const int koff = hh * 2;           // K sub-offset for A/B operand layout

    // ---------- Phase A: TDM async tile load of x[s] (32x64 f32) into LDS ----------
    if (tid < 32) {  // whole wave 0 takes branch (full EXEC); TDM issues once per wave
        unsigned long long gaddr =
            (unsigned long long)(const void*)(x + (size_t)s * 32 * NMO_);
        unsigned lds_addr = (unsigned)(unsigned long long)(void*)&lds_x[0];

        u32x4 g0;
        g0[0] = 1u;                                         // count=1, user D#
        g0[1] = lds_addr;                                   // lds_addr [63:32]
        g0[2] = (unsigned)(gaddr & 0xFFFFFFFFull);          // global_addr [95:64]
        g0[3] = (unsigned)((gaddr >> 32) & 0x01FFFFFFull)   // global_addr [120:96]
              | (2u << 30);                                 // type=2 ("image")

        i32x8 g1;
        g1[0] = (int)(2u << 16);    // workgroup_mask=0 (no cluster), data_size=2 (4B)
        g1[1] = (int)(64u << 16);   // tensor_dim0 = 64 (low16 at bits [63:48])
        g1[2] = (int)(32u << 16);   // tensor_dim0 hi=0; tensor_dim1 = 32 (bits [95:80])
        g1[3] = (int)(64u << 16);   // tensor_dim1 hi=0; tile_dim0 = 64 (bits [127:112])
        g1[4] = 32;                 // tile_dim1 = 32; tile_dim2 = 0
        g1[5] = 64;                 // tensor_dim0_stride = 64 (low 32)
        g1[6] = 0;                  // stride hi / tensor_dim1_stride (unused, 2D)
        g1[7] = 0;

        i32x4 gz4 = {0, 0, 0, 0};            // groups 2/3: tile_dim3/4 = 0 (2-D tensor)
        i32x8 gz8 = {0, 0, 0, 0, 0, 0, 0, 0};
        __builtin_amdgcn_tensor_load_to_lds(g0, g1, gz4, gz4, gz8, 0);
        __builtin_amdgcn_s_wait_tensorcnt(0);
    }
    __syncthreads();

    // ---------- Phase B: load A operands (sample rows) in WMMA A-layout ----------
    // 32-bit A 16x4: lanes 0-15 hold K=0,1 (VGPR0,1); lanes 16-31 hold K=2,3.
    v2f aUp[16], aDn[16];
#pragma unroll
    for (int k = 0; k < 16; ++k) {
        const float* pu = &lds_x[row * NMO_ + 4 * k + koff];
        aUp[k][0] = pu[0];
        aUp[k][1] = pu[1];
        const float* pd = &lds_x[(row + 16) * NMO_ + 4 * k + koff];
        aDn[k][0] = pd[0];
        aDn[k][1] = pd[1];
    }

    // ---------- Phase C: per-config gather (WMMA) + dual LU ----------
    for (int cc = 0; cc < 16; ++cc) {
        const int c = wave * 16 + cc;
        const int idxUp = cup[c * 16 + row];   // column 'row' selects orbital idxUp
        const int idxDn = cdn[c * 16 + row];

        v8f dUp = {0.f, 0.f, 0.f, 0.f, 0.f, 0.f, 0.f, 0.f};
        v8f dDn = {0.f, 0.f, 0.f, 0.f, 0.f, 0.f, 0.f, 0.f};

        // M = X(16x64) * S(64x16), S[k][n] = (idx[n]==k). 0/1 entries => exact in f32.
#pragma unroll
        for (int k = 0; k < 16; ++k) {
            const int kg = 4 * k + koff;       // this lane's K coords in B chunk
            v2f bu, bd;
            bu[0] = (idxUp == kg)     ? 1.0f : 0.0f;
            bu[1] = (idxUp == kg + 1) ? 1.0f : 0.0f;
            bd[0] = (idxDn == kg)     ? 1.0f : 0.0f;
            bd[1] = (idxDn == kg + 1) ? 1.0f : 0.0f;
            dUp = __builtin_amdgcn_wmma_f32_16x16x4_f32(
                false, aUp[k], false, bu, (short)0, dUp, false, false);
            dDn = __builtin_amdgcn_wmma_f32_16x16x4_f32(
                false, aDn[k], false, bd, (short)0, dDn, false, false);
        }

        // Scatter C/D layout (lane<16: M=r,N=lane ; lane>=16: M=r+8,N=lane-16)
        // into row-major LDS scratch (stride 17 avoids bank conflicts).
#pragma unroll
        for (int r = 0; r < 8; ++r) {
            lds_mat[wave][0][hh * 8 + r][row] = dUp[r];
            lds_mat[wave][1][hh * 8 + r][row] = dDn[r];
        }

        // Reload row-per-lane: lanes 0-15 hold up-matrix rows, 16-31 down rows.
        float m[16];
#pragma unroll
        for (int j = 0; j < 16; ++j) m[j] = lds_mat[wave][hh][row][j];

        // In-register LU without pivoting; both halves run in lockstep.
        const int bk = hh << 4;  // base lane of my 16-lane group
#pragma unroll
        for (int k = 0; k < 16; ++k) {
            const float piv  = __shfl(m[k], bk + k, 32);
            const float mult = (row > k) ? (m[k] / piv) : 0.0f;
#pragma unroll
            for (int j = 0; j < 16; ++j) {
                if (j > k) {
                    const float rkj = __shfl(m[j], bk + k, 32);
                    m[j] = fmaf(-mult, rkj, m[j]);
                }
            }
        }

        // det = product of U diagonal (broadcast each U[j][j] from lane bk+j).
        float det = 1.0f;
#pragma unroll
        for (int j = 0; j < 16; ++j) det *= __shfl(m[j], bk + j, 32);

        const float detDn = __shfl(det, 16 + row, 32);  // pull down-half det
        if (lane == 0) out[s * NCONF_ + c] = det * detDn;
    }
}

extern "C" void kernel_launch(void* const* d_in, const int* in_sizes, int n_in,
                              void* d_out, int out_size, void* d_ws, size_t ws_size,
                              hipStream_t stream) {
    const float* x   = (const float*)d_in[0];
    const int*   cup = (const int*)d_in[1];
    const int*   cdn = (const int*)d_in[2];
    float*       out = (float*)d_out;
    (void)in_sizes; (void)n_in; (void)out_size; (void)d_ws; (void)ws_size;

    slater_det_kernel<<<dim3(NSAMP_), dim3(256), 0, stream>>>(x, cup, cdn, out);
}